// Collision_65901978190203
// MI455X (gfx1250) — compile-verified
//
#include <hip/hip_runtime.h>

typedef __attribute__((ext_vector_type(2))) float v2f;
typedef __attribute__((ext_vector_type(8))) float v8f;

// Problem constants (from the reference): B=4, NV=16384, NC=8192, K=4096, D=3.
#define B_   4
#define NV_  16384
#define NC_  8192
#define K_   4096
#define COLTILES_ (K_ / 16)      // 256 column tiles of 16
#define ROWS_PER_WG_ 128         // 8 waves x 16 rows

__global__ __launch_bounds__(256) void collision_nn_wmma(
    const float* __restrict__ vertices,     // [B, NV, 3] f32
    const float* __restrict__ collider,     // [B, NC, 3] f32
    const long long* __restrict__ cv,       // [K] int64 indices into NC
    float* __restrict__ out)                // [B, NV, 2] (batch, argmin) as f32
{
    // Gathered collider panel, WMMA-B friendly: panel[k] = (-2cx, -2cy, -2cz, |c|^2)
    extern __shared__ float4 panel[];       // K_ * 16 bytes = 64 KB

    const int b = blockIdx.y;

    // ---- Stage gathered B-panel into LDS (once per workgroup) ----
    for (int k = threadIdx.x; k < K_; k += blockDim.x) {
        const long long ci = cv[k];
        const float* c = collider + ((size_t)b * NC_ + (size_t)ci) * 3;
        const float cx = c[0], cy = c[1], cz = c[2];
        panel[k] = make_float4(-2.0f * cx, -2.0f * cy, -2.0f * cz,
                               cx * cx + cy * cy + cz * cz);
    }
    __syncthreads();

    const int lane = threadIdx.x & 31;
    const int l16  = lane & 15;
    const int half = lane >> 4;                       // 0: K=0,1 | 1: K=2,3
    const int rowbase = blockIdx.x * ROWS_PER_WG_ + (threadIdx.x >> 5) * 16;

    // ---- A fragment (16x4 f32): row = (vx, vy, vz, 1) ----
    // Lanes 0-15 hold K=0 (VGPR0), K=1 (VGPR1) of row M=lane;
    // lanes 16-31 hold K=2, K=3 of row M=lane-16.
    const float* v = vertices + ((size_t)b * NV_ + (size_t)(rowbase + l16)) * 3;
    const float vx = v[0], vy = v[1], vz = v[2];
    v2f afrag;
    afrag[0] = half ? vz : vx;
    afrag[1] = half ? 1.0f : vy;

    float bestd[8];
    int   besti[8];
    #pragma unroll
    for (int r = 0; r < 8; ++r) { bestd[r] = __builtin_inff(); besti[r] = 0; }

    const v8f czero = {};

    // ---- Stream all 256 column tiles: 1 WMMA + fused argmin update each ----
    #pragma unroll 4
    for (int t = 0; t < COLTILES_; ++t) {
        const int col = t * 16 + l16;
        // B fragment (4x16 f32): lanes 0-15 -> (K=0,K=1) of col N=lane,
        // lanes 16-31 -> (K=2,K=3) of col N=lane-16.  ds_load_b64 per lane.
        const float2* p2 = reinterpret_cast<const float2*>(&panel[col]);
        const float2 bf = p2[half];
        v2f bfrag;
        bfrag[0] = bf.x;
        bfrag[1] = bf.y;

        // d[r] = c_sq - 2 v.c for rows (rowbase+r, rowbase+8+r), col = t*16 + l16
        v8f d = __builtin_amdgcn_wmma_f32_16x16x4_f32(
            /*neg_a=*/false, afrag, /*neg_b=*/false, bfrag,
            /*c_mod=*/(short)0, czero, /*reuse_a=*/false, /*reuse_b=*/false);

        #pragma unroll
        for (int r = 0; r < 8; ++r) {
            const bool take = d[r] < bestd[r];
            bestd[r] = take ? d[r] : bestd[r];
            besti[r] = take ? col  : besti[r];
        }
    }

    // ---- Cross-lane argmin reduction within each 16-lane half (branchless) ----
    // VGPR r holds row rowbase+r (lanes 0-15) and rowbase+8+r (lanes 16-31),
    // each lane covering columns == l16 (mod 16). First-occurrence tie-break.
    #pragma unroll
    for (int r = 0; r < 8; ++r) {
        float d = bestd[r];
        int   i = besti[r];
        #pragma unroll
        for (int m = 1; m <= 8; m <<= 1) {
            const float od = __shfl_xor(d, m, 32);
            const int   oi = __shfl_xor(i, m, 32);
            // bitwise | / & : no short-circuit -> pure v_cmp + v_cndmask
            const bool take = (od < d) | ((od == d) & (oi < i));
            d = take ? od : d;
            i = take ? oi : i;
        }
        if (l16 == 0) {                      // lanes 0 and 16 publish
            const int row = rowbase + r + (half << 3);
            float2* o = reinterpret_cast<float2*>(out + ((size_t)b * NV_ + (size_t)row) * 2);
            *o = make_float2((float)b, (float)i);   // one global_store_b64
        }
    }
}

extern "C" void kernel_launch(void* const* d_in, const int* in_sizes, int n_in,
                              void* d_out, int out_size, void* d_ws, size_t ws_size,
                              hipStream_t stream) {
    const float*     vertices = (const float*)d_in[0];      // [B, NV, 3] f32
    const float*     collider = (const float*)d_in[1];      // [B, NC, 3] f32
    const long long* cv       = (const long long*)d_in[2];  // [K] int64
    float*           out      = (float*)d_out;              // [B, NV, 2]

    dim3 grid(NV_ / ROWS_PER_WG_, B_);   // (128, 4)
    dim3 block(256);                     // 8 wave32s
    const size_t lds_bytes = (size_t)K_ * sizeof(float4);   // 64 KB panel

    collision_nn_wmma<<<grid, block, lds_bytes, stream>>>(vertices, collider, cv, out);
}